// PoseLoss_46557445489146
// MI455X (gfx1250) — compile-verified
//
#include <hip/hip_runtime.h>
#include <math.h>

typedef __attribute__((ext_vector_type(2))) float v2f;
typedef __attribute__((ext_vector_type(8))) float v8f;

#define B_SZ   128
#define N_PTS  1000
#define N_PAD  1008          // next multiple of 16
#define NT     (N_PAD / 16)  // 63 tiles
#define BLOCK  256           // 8 waves (wave32)
#define NWAVES (BLOCK / 32)

__global__ void poseloss_zero_out(float* out) { out[0] = 0.0f; }

__global__ __launch_bounds__(BLOCK)
void poseloss_kernel(const float* __restrict__ pred_r,   // [B,3,3]
                     const float* __restrict__ pred_t,   // [B,1,3]
                     const float* __restrict__ targets,  // [B,N,3]
                     const float* __restrict__ model,    // [B,N,3]
                     const int*   __restrict__ idxs,     // [B]
                     float* __restrict__ out)            // [1]
{
    // Pre-packed WMMA fragments, 2 floats per lane per 16-point tile, in the
    // exact 32-bit 16x4 A/B lane layout: lane<16 -> (x,y) [K=0,1],
    // lane>=16 -> (z,0) [K=2,3].
    __shared__ float sA[NT * 64];    // pred fragments
    __shared__ float sB[NT * 64];    // target fragments
    __shared__ float sp2[N_PAD];     // |pred|^2 per row
    __shared__ float st2[N_PAD];     // |tgt|^2 per col (pad = 1e30 so it never wins)
    __shared__ float red[BLOCK];

    const int b   = blockIdx.x;
    const int tid = threadIdx.x;

    const int  id  = idxs[b];
    const bool sym = (id == 0) | (id == 7) | (id == 12) | (id == 20);

    // Broadcast rotation / translation (scalar loads).
    const float* R = pred_r + b * 9;
    const float* T = pred_t + b * 3;
    const float r00 = R[0], r01 = R[1], r02 = R[2];
    const float r10 = R[3], r11 = R[4], r12 = R[5];
    const float r20 = R[6], r21 = R[7], r22 = R[8];
    const float t0 = T[0], t1 = T[1], t2c = T[2];

    const float* mp = model   + (size_t)b * N_PTS * 3;
    const float* tg = targets + (size_t)b * N_PTS * 3;

    float acc = 0.0f;

    // Stage: compute pred = R*m + t per point, pack fragments; for the
    // non-symmetric batch the per-point distance is accumulated right here.
    for (int i = tid; i < N_PAD; i += BLOCK) {
        float px = 0.f, py = 0.f, pz = 0.f;
        float qx = 0.f, qy = 0.f, qz = 0.f, q2 = 1.0e30f;
        if (i < N_PTS) {
            const float mx = mp[i*3+0], my = mp[i*3+1], mz = mp[i*3+2];
            px = r00*mx + r01*my + r02*mz + t0;
            py = r10*mx + r11*my + r12*mz + t1;
            pz = r20*mx + r21*my + r22*mz + t2c;
            qx = tg[i*3+0]; qy = tg[i*3+1]; qz = tg[i*3+2];
            q2 = qx*qx + qy*qy + qz*qz;
            if (!sym) {
                const float dx = px - qx, dy = py - qy, dz = pz - qz;
                acc += sqrtf(dx*dx + dy*dy + dz*dz);
            }
        }
        const int base = (i >> 4) * 64 + (i & 15) * 2;   // tile*64 + lane*2
        sA[base +  0] = px;  sA[base +  1] = py;         // lanes 0..15: (x,y)
        sA[base + 32] = pz;  sA[base + 33] = 0.0f;       // lanes 16..31: (z,0)
        sB[base +  0] = qx;  sB[base +  1] = qy;
        sB[base + 32] = qz;  sB[base + 33] = 0.0f;
        sp2[i] = px*px + py*py + pz*pz;
        st2[i] = q2;
    }
    __syncthreads();

    if (sym) {
        const int wid  = tid >> 5;
        const int lane = tid & 31;
        const int half = lane >> 4;
        const int lcol = lane & 15;

        for (int rt = wid; rt < NT; rt += NWAVES) {      // wave-uniform
            const v2f a = *(const v2f*)&sA[rt * 64 + lane * 2];

            float rmin[8];
            #pragma unroll
            for (int r = 0; r < 8; ++r) rmin[r] = 3.0e38f;

            for (int ct = 0; ct < NT; ++ct) {
                const v2f   bb = *(const v2f*)&sB[ct * 64 + lane * 2];
                const float c2 = st2[ct * 16 + lcol];    // t2 of this lane's column

                v8f c = {};
                // D[m][n] = pred_m . tgt_n  (16x16 tile, K=3 padded to 4)
                c = __builtin_amdgcn_wmma_f32_16x16x4_f32(
                        false, a, false, bb, (short)0, c, false, false);

                #pragma unroll
                for (int r = 0; r < 8; ++r)
                    rmin[r] = fminf(rmin[r], fmaf(-2.0f, c[r], c2));  // t2 - 2*dot
            }

            // Min across the 16 lanes (columns) of each half-wave.
            #pragma unroll
            for (int r = 0; r < 8; ++r) {
                float v = rmin[r];
                v = fminf(v, __shfl_xor(v, 1, 16));
                v = fminf(v, __shfl_xor(v, 2, 16));
                v = fminf(v, __shfl_xor(v, 4, 16));
                v = fminf(v, __shfl_xor(v, 8, 16));
                rmin[r] = v;
            }

            // Lane 0 of each half owns rows r (lower) / r+8 (upper).
            if (lcol == 0) {
                #pragma unroll
                for (int r = 0; r < 8; ++r) {
                    const int m = rt * 16 + r + 8 * half;
                    if (m < N_PTS) {
                        const float d2 = sp2[m] + rmin[r];
                        acc += sqrtf(fmaxf(d2, 0.0f));   // = ||pred - nearest tgt||
                    }
                }
            }
        }
    }

    // Block reduction, then scaled atomic accumulate into the scalar output.
    red[tid] = acc;
    __syncthreads();
    for (int s = BLOCK / 2; s > 0; s >>= 1) {
        if (tid < s) red[tid] += red[tid + s];
        __syncthreads();
    }
    if (tid == 0) {
        atomicAdd(out, red[0] * (1.0f / ((float)N_PTS * (float)B_SZ)));
    }
}

extern "C" void kernel_launch(void* const* d_in, const int* in_sizes, int n_in,
                              void* d_out, int out_size, void* d_ws, size_t ws_size,
                              hipStream_t stream) {
    (void)in_sizes; (void)n_in; (void)out_size; (void)d_ws; (void)ws_size;

    const float* pred_r  = (const float*)d_in[0];
    const float* pred_t  = (const float*)d_in[1];
    const float* targets = (const float*)d_in[2];
    const float* model   = (const float*)d_in[3];
    const int*   idxs    = (const int*)d_in[4];
    float* out = (float*)d_out;

    poseloss_zero_out<<<1, 1, 0, stream>>>(out);
    poseloss_kernel<<<B_SZ, BLOCK, 0, stream>>>(pred_r, pred_t, targets, model,
                                                idxs, out);
}